// YOLOv1Loss_55353538511504
// MI455X (gfx1250) — compile-verified
//
#include <hip/hip_runtime.h>
#include <math.h>

#define NFEAT 25
#define NCLS 20
#define SGRID 7
#define LAMBDA_COORD 5.0f
#define LAMBDA_NOOBJ 0.5f
#define EPS_SQRT 1e-6f
#define EPS_IOU 1e-10f
#define BLK 256

typedef __attribute__((ext_vector_type(2))) float v2f;
typedef __attribute__((ext_vector_type(8))) float v8f;

#if __has_builtin(__builtin_amdgcn_global_load_async_to_lds_b32) && \
    __has_builtin(__builtin_amdgcn_s_wait_asynccnt)
#define HAS_ASYNC_LDS 1
#else
#define HAS_ASYNC_LDS 0
#endif

#if __has_builtin(__builtin_amdgcn_ds_swizzle)
// XOR lane-swap within group of 32: offset = xor_mask<<10 | and_mask(0x1f)
__device__ __forceinline__ float xor_add(float v, int xmask) {
  int o;
  switch (xmask) {  // ds_swizzle offset must be an immediate
    case 8:  o = __builtin_amdgcn_ds_swizzle(__float_as_int(v), (8 << 10) | 0x1f); break;
    case 4:  o = __builtin_amdgcn_ds_swizzle(__float_as_int(v), (4 << 10) | 0x1f); break;
    case 2:  o = __builtin_amdgcn_ds_swizzle(__float_as_int(v), (2 << 10) | 0x1f); break;
    default: o = __builtin_amdgcn_ds_swizzle(__float_as_int(v), (1 << 10) | 0x1f); break;
  }
  return v + __int_as_float(o);
}
#else
__device__ __forceinline__ float xor_add(float v, int xmask) {
  return v + __shfl_xor(v, xmask, 32);
}
#endif

// ---------------------------------------------------------------------------
// Wave-level sum of 64 f32 values using V_WMMA_F32_16X16X4_F32:
//   D = ones(16x4) x B(4x16) + 0  ->  every row of D holds the 16 column sums
// of B. Any (k,j)<->slot bijection works since we sum everything; reduces
// 64 -> 16 in one matrix op with a fixed hardware summation order
// (deterministic). Remaining 16 -> 1 via XOR lane swaps (stays within the
// low-16 lane group, so the duplicated rows in lanes 16..31 are masked out).
// Must be executed by a fully-active wave32 (EXEC all ones).
// ---------------------------------------------------------------------------
__device__ __forceinline__ float wave_sum64(const float* s64, int lane) {
#if __has_builtin(__builtin_amdgcn_wmma_f32_16x16x4_f32)
  v2f a; a[0] = 1.0f; a[1] = 1.0f;                 // A = ones (all slots)
  v2f b; b[0] = s64[lane]; b[1] = s64[lane + 32];  // B = 64 partials
  v8f c = {0.f, 0.f, 0.f, 0.f, 0.f, 0.f, 0.f, 0.f};
  c = __builtin_amdgcn_wmma_f32_16x16x4_f32(
      /*neg_a=*/false, a, /*neg_b=*/false, b,
      /*c_mod=*/(short)0, c, /*reuse_a=*/false, /*reuse_b=*/false);
  // D VGPR0: lanes 0..15 = row M=0 (col sums); lanes 16..31 = row M=8 (same
  // values -> zero them to avoid double counting).
  float v = (lane < 16) ? c[0] : 0.0f;
#else
  float v = s64[lane] + s64[lane + 32];
  v += __shfl_xor(v, 16, 32);
#endif
  v = xor_add(v, 8);
  v = xor_add(v, 4);
  v = xor_add(v, 2);
  v = xor_add(v, 1);
  return v;  // total valid in lane 0
}

// Deterministic 256-thread block reduction (fixed tree order), result @ tid 0.
__device__ __forceinline__ float block_reduce_256(float local, float* sdata) {
  const int tid = threadIdx.x;
  sdata[tid] = local;
  __syncthreads();
  if (tid < 128) sdata[tid] += sdata[tid + 128];
  __syncthreads();
  if (tid < 64) sdata[tid] += sdata[tid + 64];
  __syncthreads();
  float r = 0.f;
  if (tid < 32) r = wave_sum64(sdata, tid);  // wave 0 fully active
  return r;
}

__global__ void k_init(int* cnt) {
  if (threadIdx.x == 0) *cnt = 0;
}

// Pass 1: per-cell coord + class + noobj-conf losses; compact object targets.
__global__ void k_prep(const float* __restrict__ p, const float* __restrict__ t,
                       int M, int* __restrict__ cnt,
                       float* __restrict__ gx1, float* __restrict__ gy1,
                       float* __restrict__ gx2, float* __restrict__ gy2,
                       float* __restrict__ ga, float* __restrict__ partials) {
  __shared__ float sred[BLK];
  const int tid = threadIdx.x;
  const int i = blockIdx.x * BLK + tid;
  float local = 0.f;
  if (i < M) {
    const float* pi = p + (size_t)i * NFEAT;
    const float* ti = t + (size_t)i * NFEAT;
    const float t4 = ti[4];
    const float p4 = pi[4];
    if (t4 > 0.f) {  // object cell
      const float dx = pi[0] - ti[0];
      const float dy = pi[1] - ti[1];
      const float dw = sqrtf(pi[2] + EPS_SQRT) - sqrtf(ti[2] + EPS_SQRT);
      const float dh = sqrtf(pi[3] + EPS_SQRT) - sqrtf(ti[3] + EPS_SQRT);
      local += LAMBDA_COORD * (dx * dx + dy * dy + dw * dw + dh * dh);
#pragma unroll
      for (int c = 0; c < NCLS; ++c) {
        const float d = pi[5 + c] - ti[5 + c];
        local += d * d;
      }
      // Compact this target's corners/area. Slot order is nondeterministic,
      // but the only consumer is a max() over the set -> order-invariant.
      const int slot = atomicAdd(cnt, 1);
      const float x1 = ti[0] - 0.5f * ti[2], x2 = ti[0] + 0.5f * ti[2];
      const float y1 = ti[1] - 0.5f * ti[3], y2 = ti[1] + 0.5f * ti[3];
      gx1[slot] = x1; gy1[slot] = y1; gx2[slot] = x2; gy2[slot] = y2;
      ga[slot] = (x2 - x1) * (y2 - y1);
    }
    if (t4 == 0.f) {  // empty cell
      const float d = p4 - t4;
      local += LAMBDA_NOOBJ * d * d;
    }
  }
  const float r = block_reduce_256(local, sred);
  if (tid == 0) partials[blockIdx.x] = r;
}

// Pass 2: for object rows only, conf_t = max IoU vs compacted targets staged
// into LDS (async DMA path on gfx1250), accumulate (p_conf - conf_t)^2.
__global__ void k_conf(const float* __restrict__ p, const float* __restrict__ t,
                       int M, const int* __restrict__ cnt,
                       const float* __restrict__ gx1, const float* __restrict__ gy1,
                       const float* __restrict__ gx2, const float* __restrict__ gy2,
                       const float* __restrict__ ga, float* __restrict__ partials) {
  __shared__ float sx1[BLK], sy1[BLK], sx2[BLK], sy2[BLK], sa[BLK];
  __shared__ float sred[BLK];
  const int tid = threadIdx.x;
  const int i = blockIdx.x * BLK + tid;
  const int K = *cnt;

  bool active = false;
  float px1 = 0.f, py1 = 0.f, px2 = 0.f, py2 = 0.f, pa = 0.f, p4 = 0.f;
  float maxv = -1.f;  // matches reference fill of -1 for masked columns / K==0
  if (i < M && t[(size_t)i * NFEAT + 4] > 0.f) {
    active = true;
    const float* pi = p + (size_t)i * NFEAT;
    px1 = pi[0] - 0.5f * pi[2]; px2 = pi[0] + 0.5f * pi[2];
    py1 = pi[1] - 0.5f * pi[3]; py2 = pi[1] + 0.5f * pi[3];
    pa = (px2 - px1) * (py2 - py1);
    p4 = pi[4];
  }

  for (int j0 = 0; j0 < K; j0 += BLK) {  // K is block-uniform: barriers safe
    const int nj = min(BLK, K - j0);
    __syncthreads();
#if HAS_ASYNC_LDS
    if (tid < nj) {
      // GLOBAL_LOAD_ASYNC_TO_LDS_B32: memory -> LDS without VGPR round-trip;
      // tracked by ASYNCcnt, drained below before the barrier.
      // Builtin signature (per hipcc diagnostic): (int* src, int* dst, Ii, Ii)
      __builtin_amdgcn_global_load_async_to_lds_b32(
          (int*)(gx1 + j0 + tid), (int*)(sx1 + tid), 0, 0);
      __builtin_amdgcn_global_load_async_to_lds_b32(
          (int*)(gy1 + j0 + tid), (int*)(sy1 + tid), 0, 0);
      __builtin_amdgcn_global_load_async_to_lds_b32(
          (int*)(gx2 + j0 + tid), (int*)(sx2 + tid), 0, 0);
      __builtin_amdgcn_global_load_async_to_lds_b32(
          (int*)(gy2 + j0 + tid), (int*)(sy2 + tid), 0, 0);
      __builtin_amdgcn_global_load_async_to_lds_b32(
          (int*)(ga + j0 + tid), (int*)(sa + tid), 0, 0);
    }
    __builtin_amdgcn_s_wait_asynccnt(0);
    __syncthreads();
#else
    if (tid < nj) {
      sx1[tid] = gx1[j0 + tid]; sy1[tid] = gy1[j0 + tid];
      sx2[tid] = gx2[j0 + tid]; sy2[tid] = gy2[j0 + tid];
      sa[tid] = ga[j0 + tid];
    }
    __syncthreads();
#endif
    if (active) {
      for (int j = 0; j < nj; ++j) {
        float iw = fminf(px2, sx2[j]) - fmaxf(px1, sx1[j]);
        float ih = fminf(py2, sy2[j]) - fmaxf(py1, sy1[j]);
        iw = fmaxf(iw, 0.f);
        ih = fmaxf(ih, 0.f);
        const float inter = iw * ih;
        const float iou = inter / (pa + sa[j] - inter + EPS_IOU);
        maxv = fmaxf(maxv, iou);
      }
    }
  }

  float local = 0.f;
  if (active) {
    const float d = p4 - maxv;
    local = d * d;
  }
  __syncthreads();
  const float r = block_reduce_256(local, sred);
  if (tid == 0) partials[blockIdx.x] = r;
}

// Final: one wave sums np block partials (deterministic per-lane strided
// accumulation + WMMA 64->1), scales by 1/B.
__global__ void k_final(const float* __restrict__ partials, int np, float invB,
                        float* __restrict__ out) {
  __shared__ float s64[64];
  const int lane = threadIdx.x;  // blockDim == 32
  float s0 = 0.f, s1 = 0.f;
  for (int j = lane; j < np; j += 64) s0 += partials[j];
  for (int j = lane + 32; j < np; j += 64) s1 += partials[j];
  s64[lane] = s0;
  s64[lane + 32] = s1;
  __syncthreads();
  const float v = wave_sum64(s64, lane);
  if (lane == 0) out[0] = v * invB;
}

extern "C" void kernel_launch(void* const* d_in, const int* in_sizes, int n_in,
                              void* d_out, int out_size, void* d_ws, size_t ws_size,
                              hipStream_t stream) {
  const float* pred = (const float*)d_in[0];
  const float* target = (const float*)d_in[1];
  const int M = in_sizes[0] / NFEAT;                     // B*S*S
  const float invB = (float)(SGRID * SGRID) / (float)M;  // 1/B

  char* ws = (char*)d_ws;
  int* cnt = (int*)ws;
  float* gx1 = (float*)(ws + 16);
  float* gy1 = gx1 + M;
  float* gx2 = gy1 + M;
  float* gy2 = gx2 + M;
  float* ga  = gy2 + M;
  float* partials = ga + M;  // 2*NB floats

  const int NB = (M + BLK - 1) / BLK;

  k_init<<<1, 32, 0, stream>>>(cnt);
  k_prep<<<NB, BLK, 0, stream>>>(pred, target, M, cnt, gx1, gy1, gx2, gy2, ga,
                                 partials);
  k_conf<<<NB, BLK, 0, stream>>>(pred, target, M, cnt, gx1, gy1, gx2, gy2, ga,
                                 partials + NB);
  k_final<<<1, 32, 0, stream>>>(partials, 2 * NB, invB, (float*)d_out);
}